// ClassificationHead_34686155882985
// MI455X (gfx1250) — compile-verified
//
#include <hip/hip_runtime.h>
#include <hip/hip_bf16.h>

typedef float __attribute__((ext_vector_type(2))) v2f;
typedef float __attribute__((ext_vector_type(8))) v8f;

#define T_TASKS 256
#define NQ      300
#define NS      25
#define D       640
#define NWAY    5
#define LAMBDA  50.0f

// ---------------------------------------------------------------------------
// Kernel 1: per task, compute K = S S^T + lambda I, solve K X = 2 Y
// (Gauss-Jordan, SPD + 50I so no pivoting needed), then Wmat = S^T X (640x5).
// One block (256 threads) per task. S staged through LDS in two 320-col
// chunks with stride-321 padding (bank-conflict free).
// ---------------------------------------------------------------------------
#define CHUNK   320
#define SSTRIDE 321

__global__ void ridge_prep_kernel(const float* __restrict__ support,
                                  const int* __restrict__ labels,
                                  float* __restrict__ Wg) {
    const int task = blockIdx.x;
    const int tid  = threadIdx.x;

    __shared__ float Sl[NS * SSTRIDE];     // 32.1 KB
    __shared__ float Amat[NS * 32];        // augmented [K | 2Y], 32-col stride
    __shared__ float fac[NS];
    __shared__ float Xl[NS * NWAY];

    const float* sg = support + (size_t)task * NS * D;

    // init K accumulator
    for (int idx = tid; idx < NS * NS; idx += 256) {
        Amat[(idx / NS) * 32 + (idx % NS)] = 0.0f;
    }
    __syncthreads();

    // K = S S^T, accumulated over two k-chunks
    for (int c0 = 0; c0 < D; c0 += CHUNK) {
        for (int idx = tid; idx < NS * CHUNK; idx += 256) {
            int r = idx / CHUNK, k = idx - r * CHUNK;
            Sl[r * SSTRIDE + k] = sg[r * D + c0 + k];
        }
        __syncthreads();
        for (int idx = tid; idx < NS * NS; idx += 256) {
            int i = idx / NS, j = idx - i * NS;
            float acc = 0.0f;
            #pragma unroll 4
            for (int k = 0; k < CHUNK; ++k)
                acc += Sl[i * SSTRIDE + k] * Sl[j * SSTRIDE + k];
            Amat[i * 32 + j] += acc;
        }
        __syncthreads();
    }

    // + lambda I, RHS = 2*onehot(labels)
    if (tid < NS) Amat[tid * 32 + tid] += LAMBDA;
    if (tid < NS * NWAY) Amat[(tid / NWAY) * 32 + NS + (tid % NWAY)] = 0.0f;
    __syncthreads();
    if (tid < NS) {
        int lab = labels[task * NS + tid];
        Amat[tid * 32 + NS + lab] = 2.0f;
    }
    __syncthreads();

    // Gauss-Jordan elimination on [K | 2Y]
    for (int p = 0; p < NS; ++p) {
        if (tid < NS) fac[tid] = Amat[tid * 32 + p] / Amat[p * 32 + p];
        __syncthreads();
        for (int idx = tid; idx < NS * 32; idx += 256) {
            int r = idx >> 5, c = idx & 31;
            if (r != p && c > p && c < NS + NWAY)
                Amat[r * 32 + c] -= fac[r] * Amat[p * 32 + c];
        }
        __syncthreads();
    }
    // X = diag(K)^-1 * rhs
    if (tid < NS * NWAY) {
        int i = tid / NWAY, w = tid - i * NWAY;
        Xl[tid] = Amat[i * 32 + NS + w] / Amat[i * 32 + i];
    }
    __syncthreads();

    // Wmat[k][w] = sum_s S[s][k] * X[s][w]   (640 x 5), chunked S reload
    float* wout = Wg + (size_t)task * D * NWAY;
    for (int c0 = 0; c0 < D; c0 += CHUNK) {
        for (int idx = tid; idx < NS * CHUNK; idx += 256) {
            int r = idx / CHUNK, k = idx - r * CHUNK;
            Sl[r * SSTRIDE + k] = sg[r * D + c0 + k];
        }
        __syncthreads();
        for (int idx = tid; idx < CHUNK * NWAY; idx += 256) {
            int kk = idx / NWAY, w = idx - kk * NWAY;
            float acc = 0.0f;
            #pragma unroll
            for (int s = 0; s < NS; ++s)
                acc += Sl[s * SSTRIDE + kk] * Xl[s * NWAY + w];
            wout[(c0 + kk) * NWAY + w] = acc;
        }
        __syncthreads();
    }
}

// ---------------------------------------------------------------------------
// Kernel 2: logits = scale * Q @ Wmat using V_WMMA_F32_16X16X4_F32.
// grid = (5 m-groups of 64 rows, 256 tasks), block = 128 threads (4 waves).
// Each wave owns one 16-row query tile; Wmat lives in LDS as a 640x16
// zero-padded fp32 image (B-matrix layout per 4-row k-slice).
// ---------------------------------------------------------------------------
__global__ void ridge_logits_kernel(const float* __restrict__ query,
                                    const float* __restrict__ Wg,
                                    const float* __restrict__ scale,
                                    float* __restrict__ out) {
    const int mgroup = blockIdx.x;       // 0..4  (64 query rows each)
    const int task   = blockIdx.y;       // 0..255
    const int tid    = threadIdx.x;      // 0..127

    __shared__ float Wlds[D * 16];       // 40 KB, cols 5..15 zero

    const float* wsrc = Wg + (size_t)task * D * NWAY;
    for (int idx = tid; idx < D * 16; idx += 128) {
        int k = idx >> 4, n = idx & 15;
        Wlds[idx] = (n < NWAY) ? wsrc[k * NWAY + n] : 0.0f;
    }
    __syncthreads();

    const int wave  = tid >> 5;          // 0..3
    const int lane  = tid & 31;
    const int half  = lane >> 4;         // 0 or 1
    const int l16   = lane & 15;
    const int mbase = mgroup * 64 + wave * 16;

    // A-matrix row for this lane (clamped; out-of-range rows masked at store)
    int m = mbase + l16;
    if (m > NQ - 1) m = NQ - 1;
    const float* qrow = query + ((size_t)task * NQ + m) * D + half * 2;

    v8f acc = {};
    #pragma unroll 8
    for (int k0 = 0; k0 < D; k0 += 4) {
        // A 16x4 tile: lane<16 -> Q[m][k0],Q[m][k0+1]; lane>=16 -> k0+2,k0+3
        v2f a = *(const v2f*)(qrow + k0);
        // B 4x16 tile: vgpr0 = rows k0 / k0+2, vgpr1 = rows k0+1 / k0+3
        int kb = k0 + 2 * half;
        v2f b;
        b.x = Wlds[kb * 16 + l16];
        b.y = Wlds[(kb + 1) * 16 + l16];
        acc = __builtin_amdgcn_wmma_f32_16x16x4_f32(
            /*neg_a=*/false, a, /*neg_b=*/false, b,
            /*c_mod=*/(short)0, acc, /*reuse_a=*/false, /*reuse_b=*/false);
    }

    const float sc = scale[0];
    // C/D layout: VGPR v -> M = mbase + v + 8*half, N = l16
    float* obase = out + (size_t)task * NQ * NWAY;
    #pragma unroll
    for (int v = 0; v < 8; ++v) {
        int row = mbase + v + half * 8;
        int col = l16;
        if (col < NWAY && row < NQ)
            obase[row * NWAY + col] = acc[v] * sc;
    }
}

extern "C" void kernel_launch(void* const* d_in, const int* in_sizes, int n_in,
                              void* d_out, int out_size, void* d_ws, size_t ws_size,
                              hipStream_t stream) {
    const float* query   = (const float*)d_in[0];   // (256, 300, 640) f32
    const float* support = (const float*)d_in[1];   // (256, 25, 640)  f32
    const float* scale   = (const float*)d_in[2];   // (1,)            f32
    const int*   labels  = (const int*)d_in[3];     // (256, 25)       i32
    (void)in_sizes; (void)n_in; (void)out_size; (void)ws_size;

    float* Wg  = (float*)d_ws;                      // 256*640*5 f32 = 3.3 MB
    float* out = (float*)d_out;                     // (256, 300, 5) f32

    ridge_prep_kernel<<<dim3(T_TASKS), dim3(256), 0, stream>>>(support, labels, Wg);
    ridge_logits_kernel<<<dim3(5, T_TASKS), dim3(128), 0, stream>>>(query, Wg, scale, out);
}